// MnistAdditionModule_5677946766055
// MI455X (gfx1250) — compile-verified
//
#include <hip/hip_runtime.h>
#include <math.h>

typedef float v2f __attribute__((ext_vector_type(2)));
typedef float v8f __attribute__((ext_vector_type(8)));

#define NIMG  16384
#define BATCH 4096
#define NSUMS 199
#define K2P   152   // conv2 im2col K (150) padded to multiple of 4

// ---------------------------------------------------------------------------
// Zero-pad FC weights so WMMA B-tiles need no predication.
// W1p: [128][256] (120 valid rows), W2p: [96][120] (84 valid), W3p: [16][84] (10 valid)
// ---------------------------------------------------------------------------
__global__ __launch_bounds__(256) void prep_weights_kernel(
    const float* __restrict__ f1w, const float* __restrict__ f2w,
    const float* __restrict__ f3w,
    float* __restrict__ W1p, float* __restrict__ W2p, float* __restrict__ W3p) {
  int i = blockIdx.x * blockDim.x + threadIdx.x;
  if (i < 128 * 256) W1p[i] = (i / 256 < 120) ? f1w[i] : 0.0f;
  if (i < 96 * 120)  W2p[i] = (i / 120 < 84)  ? f2w[i] : 0.0f;
  if (i < 16 * 84)   W3p[i] = (i / 84  < 10)  ? f3w[i] : 0.0f;
}

// ---------------------------------------------------------------------------
// Fused LeNet conv stage, one image per 128-thread block (4 waves).
//   phase 1 (VALU): conv1(5x5,1ch)+maxpool2+relu -> LDS s1[6][12][12]
//   phase 2 (WMMA): conv2 as im2col GEMM, M=16 oc, K=152(pad), N=64 positions.
//     A = zero-padded conv2 weights [16][152] (LDS)
//     B = transposed im2col Bt[n][k] (LDS) so each frag is one b64 LDS read
//     each wave owns one 16-wide N tile -> 38 wmma k-steps
//   phase 3 (VALU): bias already added; 2x2 maxpool + relu -> x2[img][256]
// Flatten order oc*16 + oy*4 + ox matches reference reshape.
// ---------------------------------------------------------------------------
__global__ __launch_bounds__(128) void lenet_conv_kernel(
    const float* __restrict__ images,
    const float* __restrict__ c1w_g, const float* __restrict__ c1b_g,
    const float* __restrict__ c2w_g, const float* __restrict__ c2b_g,
    float* __restrict__ x2) {
  __shared__ float simg[28 * 28];
  __shared__ float s1[6 * 12 * 12];
  __shared__ float c1w[150];
  __shared__ float c1b[6];
  __shared__ float c2wp[16 * K2P];   // A matrix, zero-padded k=150,151
  __shared__ float c2b[16];
  __shared__ float Bt[64 * K2P];     // im2col, [n][k]
  __shared__ float c2out[16 * 64];   // conv2 result pre-pool

  const int tid = threadIdx.x;
  const int img = blockIdx.x;

  for (int i = tid; i < 784; i += 128) simg[i] = images[(size_t)img * 784 + i];
  for (int i = tid; i < 150; i += 128) c1w[i] = c1w_g[i];
  for (int i = tid; i < 16 * K2P; i += 128) {
    const int oc = i / K2P, k = i % K2P;
    c2wp[i] = (k < 150) ? c2w_g[oc * 150 + k] : 0.0f;
  }
  if (tid < 6)  c1b[tid] = c1b_g[tid];
  if (tid < 16) c2b[tid] = c2b_g[tid];
  __syncthreads();

  // ---- phase 1: conv1 + pool + relu (VALU, 864 outputs) ----
  for (int idx = tid; idx < 864; idx += 128) {
    const int oc = idx / 144, r = idx % 144, oy = r / 12, ox = r % 12;
    float m = -INFINITY;
    for (int py = 0; py < 2; ++py) {
      for (int px = 0; px < 2; ++px) {
        const int iy = oy * 2 + py, ix = ox * 2 + px;
        float s = c1b[oc];
#pragma unroll
        for (int ky = 0; ky < 5; ++ky)
#pragma unroll
          for (int kx = 0; kx < 5; ++kx)
            s += simg[(iy + ky) * 28 + ix + kx] * c1w[oc * 25 + ky * 5 + kx];
        m = fmaxf(m, s);
      }
    }
    s1[idx] = fmaxf(m, 0.0f);
  }
  __syncthreads();

  // ---- build transposed im2col: Bt[n*152 + k] ----
  for (int i = tid; i < 64 * K2P; i += 128) {
    const int n = i / K2P, k = i % K2P;
    float v = 0.0f;
    if (k < 150) {
      const int ic = k / 25, kk = k % 25, ky = kk / 5, kx = kk % 5;
      const int py = n / 8, px = n % 8;
      v = s1[ic * 144 + (py + ky) * 12 + (px + kx)];
    }
    Bt[i] = v;
  }
  __syncthreads();

  // ---- phase 2: conv2 GEMM on the matrix pipe ----
  {
    const int lane = tid & 31;
    const int nt   = tid >> 5;            // wave id = N tile (0..3)
    const int l16  = lane & 15;
    const int koff = (lane >> 4) * 2;
    const int n0   = nt * 16;

    const float* arow = &c2wp[l16 * K2P + koff];
    const float* brow = &Bt[(n0 + l16) * K2P + koff];

    v8f acc = {0.f, 0.f, 0.f, 0.f, 0.f, 0.f, 0.f, 0.f};
    for (int k0 = 0; k0 < K2P; k0 += 4) {
      v2f a = *(const v2f*)(arow + k0);
      v2f b = *(const v2f*)(brow + k0);
      acc = __builtin_amdgcn_wmma_f32_16x16x4_f32(
          false, a, false, b, (short)0, acc, false, false);
    }
    const int hbase = (lane >> 4) << 3;   // 0 or 8
#pragma unroll
    for (int r = 0; r < 8; ++r) {
      const int oc = r + hbase;
      c2out[oc * 64 + n0 + l16] = acc[r] + c2b[oc];
    }
  }
  __syncthreads();

  // ---- phase 3: 2x2 maxpool + relu -> x2 ----
  for (int o = tid; o < 256; o += 128) {
    const int oc = o / 16, r = o % 16, oy = r / 4, ox = r % 4;
    float m = -INFINITY;
#pragma unroll
    for (int py = 0; py < 2; ++py)
#pragma unroll
      for (int px = 0; px < 2; ++px)
        m = fmaxf(m, c2out[oc * 64 + (oy * 2 + py) * 8 + (ox * 2 + px)]);
    x2[(size_t)img * 256 + o] = fmaxf(m, 0.0f);
  }
}

// ---------------------------------------------------------------------------
// FC layer as WMMA f32 16x16x4 GEMM:  Y[M,Nvalid] = relu?(X[M,K] * Wp^T + b)
// Wp is [ntiles*16][K] zero-padded row-major.  One wave per 16-row M tile,
// sweeping all N tiles.  EXEC all-ones around WMMA.
// ---------------------------------------------------------------------------
__global__ __launch_bounds__(256) void wmma_fc_kernel(
    const float* __restrict__ X, const float* __restrict__ Wp,
    const float* __restrict__ bias, float* __restrict__ Y,
    int K, int ntiles, int Nvalid, int ldY, int do_relu) {
  const int lane = threadIdx.x & 31;
  const int wid  = (blockIdx.x * blockDim.x + threadIdx.x) >> 5;
  const int m0   = wid * 16;
  const int l16  = lane & 15;
  const int koff = (lane >> 4) * 2;

  const float* arow = X + (size_t)(m0 + l16) * K + koff;

  for (int nt = 0; nt < ntiles; ++nt) {
    const float* brow = Wp + (size_t)(nt * 16 + l16) * K + koff;
    v8f acc = {0.f, 0.f, 0.f, 0.f, 0.f, 0.f, 0.f, 0.f};
    for (int k0 = 0; k0 < K; k0 += 4) {
      v2f a = *(const v2f*)(arow + k0);
      v2f b = *(const v2f*)(brow + k0);
      acc = __builtin_amdgcn_wmma_f32_16x16x4_f32(
          false, a, false, b, (short)0, acc, false, false);
    }
    const int col = nt * 16 + l16;
    if (col < Nvalid) {
      const float bv = bias[col];
      const int rbase = m0 + ((lane >> 4) << 3);
#pragma unroll
      for (int r = 0; r < 8; ++r) {
        float v = acc[r] + bv;
        if (do_relu) v = fmaxf(v, 0.0f);
        Y[(size_t)(rbase + r) * ldY + col] = v;
      }
    }
  }
}

// ---------------------------------------------------------------------------
// Circuit: per-batch block.  log_softmax over 10 logits for each of 4 digits,
// lp1/lp2 outer sums (100 each), then 199-way segment logsumexp over i+j=s.
// logits is [16384][16] padded (cols >= 10 never read).
// ---------------------------------------------------------------------------
__global__ __launch_bounds__(256) void circuit_kernel(
    const float* __restrict__ logits, float* __restrict__ out) {
  __shared__ float lp[4][10];
  __shared__ float lp1[100];
  __shared__ float lp2[100];
  const int b = blockIdx.x;
  const int tid = threadIdx.x;

  if (tid < 4) {
    const float* row = logits + (size_t)(b * 4 + tid) * 16;
    float m = row[0];
#pragma unroll
    for (int c = 1; c < 10; ++c) m = fmaxf(m, row[c]);
    float s = 0.0f;
#pragma unroll
    for (int c = 0; c < 10; ++c) s += expf(row[c] - m);
    const float lse = m + logf(s);
#pragma unroll
    for (int c = 0; c < 10; ++c) lp[tid][c] = row[c] - lse;
  }
  __syncthreads();

  if (tid < 100) {
    const int hi = tid / 10, lo = tid % 10;
    lp1[tid] = lp[0][hi] + lp[1][lo];
    lp2[tid] = lp[2][hi] + lp[3][lo];
  }
  __syncthreads();

  if (tid < NSUMS) {
    const int s = tid;
    const int i0 = (s - 99 > 0) ? s - 99 : 0;
    const int i1 = (s < 99) ? s : 99;
    float m = -INFINITY;
    for (int i = i0; i <= i1; ++i) m = fmaxf(m, lp1[i] + lp2[s - i]);
    float acc = 0.0f;
    for (int i = i0; i <= i1; ++i) acc += expf(lp1[i] + lp2[s - i] - m);
    out[(size_t)b * NSUMS + s] = m + logf(acc);
  }
}

// ---------------------------------------------------------------------------
extern "C" void kernel_launch(void* const* d_in, const int* in_sizes, int n_in,
                              void* d_out, int out_size, void* d_ws, size_t ws_size,
                              hipStream_t stream) {
  const float* images = (const float*)d_in[0];
  const float* c1w = (const float*)d_in[1];
  const float* c1b = (const float*)d_in[2];
  const float* c2w = (const float*)d_in[3];
  const float* c2b = (const float*)d_in[4];
  const float* f1w = (const float*)d_in[5];
  const float* f1b = (const float*)d_in[6];
  const float* f2w = (const float*)d_in[7];
  const float* f2b = (const float*)d_in[8];
  const float* f3w = (const float*)d_in[9];
  const float* f3b = (const float*)d_in[10];

  float* ws  = (float*)d_ws;
  float* x2  = ws;                       // [16384][256]
  float* h1  = x2  + (size_t)NIMG * 256; // [16384][120]
  float* h2  = h1  + (size_t)NIMG * 120; // [16384][84]
  float* lg  = h2  + (size_t)NIMG * 84;  // [16384][16] padded logits
  float* W1p = lg  + (size_t)NIMG * 16;  // [128][256]
  float* W2p = W1p + 128 * 256;          // [96][120]
  float* W3p = W2p + 96 * 120;           // [16][84]
  (void)in_sizes; (void)n_in; (void)out_size; (void)ws_size;

  prep_weights_kernel<<<dim3(128), dim3(256), 0, stream>>>(f1w, f2w, f3w, W1p, W2p, W3p);
  lenet_conv_kernel<<<dim3(NIMG), dim3(128), 0, stream>>>(images, c1w, c1b, c2w, c2b, x2);

  // 16384 rows -> 1024 M-tiles; 8 waves/block -> 128 blocks, exact cover.
  wmma_fc_kernel<<<dim3(128), dim3(256), 0, stream>>>(x2, W1p, f1b, h1, 256, 8, 120, 120, 1);
  wmma_fc_kernel<<<dim3(128), dim3(256), 0, stream>>>(h1, W2p, f2b, h2, 120, 6, 84, 84, 1);
  wmma_fc_kernel<<<dim3(128), dim3(256), 0, stream>>>(h2, W3p, f3b, lg, 84, 1, 10, 16, 0);

  circuit_kernel<<<dim3(BATCH), dim3(256), 0, stream>>>(lg, (float*)d_out);
}